// PatchAugmenter_53317724013214
// MI455X (gfx1250) — compile-verified
//
#include <hip/hip_runtime.h>
#include <stdint.h>

// PatchAugmenter for MI455X (gfx1250):
//   - one workgroup (1024 threads / 32 wave32s) per patch
//   - 128 KB LDS per workgroup (CDNA5 WGP has 320 KB): smA = staged source
//     plane, smB = rotation-pass intermediate plane
//   - global->LDS staging uses CDNA5 async-copy (global_load_async_to_lds_b128,
//     ASYNCcnt + s_wait_asynccnt), double-buffered against compute
//   - all resampling gathers are ds_load; all stores are 128-bit vectorized
// Traffic: ~67 MB read + ~67 MB written => ~6 us floor at 23.3 TB/s.

#define TILE 16384   // H*W = 128*128
#define HW_MASK 127

__device__ __forceinline__ void issue_stage_async(unsigned lds_byte_off,
                                                  const float* __restrict__ gsrc,
                                                  int tid) {
  // Copy 16384 floats (64 KB) global -> LDS. 1024 threads * 4 iterations,
  // one 16-byte async transfer per lane per iteration (ISA 15.18.3 op 98,
  // GV mode: each lane copies 16 bytes from global vaddr to LDS at vdst). Tracked by ASYNCcnt.
#pragma unroll
  for (int i = 0; i < 4; ++i) {
    int e = (i << 10) + tid;                                  // float4 index
    unsigned lofs = lds_byte_off + ((unsigned)e << 4);
    unsigned long long ga = (unsigned long long)(uintptr_t)(gsrc + ((size_t)e << 2));
    asm volatile("global_load_async_to_lds_b128 %0, %1, off"
                 :: "v"(lofs), "v"(ga)
                 : "memory");
  }
}

__device__ __forceinline__ void wait_async_and_barrier() {
#if __has_builtin(__builtin_amdgcn_s_wait_asynccnt)
  __builtin_amdgcn_s_wait_asynccnt(0);
#else
  asm volatile("s_wait_asynccnt 0x0" ::: "memory");
#endif
  __syncthreads();
}

// Bilinear sample of the rotation grid at intermediate pixel (h2, w2),
// reading the staged source plane in LDS. Mirrors reference _grid_sample
// ('bilinear') exactly: corner order (x0,y0),(x0+1,y0),(x0,y0+1),(x0+1,y0+1),
// value * (weight * in_bounds).
__device__ __forceinline__ float rot_bilinear(const float* __restrict__ sm,
                                              float cA, float sA,
                                              int h2, int w2) {
  const float k = 1.0f / 128.0f;
  float X = (2.0f * (float)w2 + 1.0f) * k - 1.0f;
  float Y = (2.0f * (float)h2 + 1.0f) * k - 1.0f;
  float gx = cA * X - sA * Y;           // theta_rot row 0: [c, -s, 0]
  float gy = sA * X + cA * Y;           // theta_rot row 1: [s,  c, 0]
  float ix = ((gx + 1.0f) * 128.0f - 1.0f) * 0.5f;
  float iy = ((gy + 1.0f) * 128.0f - 1.0f) * 0.5f;
  float x0 = floorf(ix), y0 = floorf(iy);
  float wx1 = ix - x0, wy1 = iy - y0;
  float wx0 = 1.0f - wx1, wy0 = 1.0f - wy1;
  float acc = 0.0f;
#pragma unroll
  for (int cy = 0; cy < 2; ++cy) {
    float yf = y0 + (float)cy;
    float wy = cy ? wy1 : wy0;
#pragma unroll
    for (int cx = 0; cx < 2; ++cx) {
      float xf = x0 + (float)cx;
      float wx = cx ? wx1 : wx0;
      float inb = (xf >= 0.0f && xf <= 127.0f && yf >= 0.0f && yf <= 127.0f)
                      ? 1.0f : 0.0f;
      int xc = (int)fminf(fmaxf(xf, 0.0f), 127.0f);
      int yc = (int)fminf(fmaxf(yf, 0.0f), 127.0f);
      acc += sm[(yc << 7) + xc] * (wx * wy * inb);
    }
  }
  return acc;
}

__global__ void __launch_bounds__(1024)
patch_augment_fused(const float* __restrict__ patches,
                    const float* __restrict__ labels,
                    const float* __restrict__ angles,
                    const uint8_t* __restrict__ flip_h,
                    const uint8_t* __restrict__ flip_v,
                    const float* __restrict__ scales,
                    float* __restrict__ out_p,
                    float* __restrict__ out_l) {
  __shared__ float smA[TILE];   // staged source plane (64 KB)
  __shared__ float smB[TILE];   // rotation intermediate plane (64 KB)

  const int n = blockIdx.x;     // patch index 0..N-1
  const int tid = threadIdx.x;  // 0..1023
  const float k = 1.0f / 128.0f;

  const float ang = angles[n];
  const float scl = scales[n];
  const int fhm = flip_h[n] ? HW_MASK : 0;  // x ^ 127 == 127 - x for 7-bit x
  const int fvm = flip_v[n] ? HW_MASK : 0;
  float sA, cA;
  sincosf(ang, &sA, &cA);

  const unsigned ldsA = (unsigned)(uintptr_t)smA;
  const float* src0 = patches + (size_t)n * 3 * TILE;
  const float* lsrc = labels + (size_t)n * TILE;

  __builtin_prefetch(lsrc, 0, 1);            // global_prefetch_b8 (L2 warm)

  issue_stage_async(ldsA, src0, tid);        // stage channel 0
  wait_async_and_barrier();

  for (int c = 0; c < 3; ++c) {
    // ---- phase A: rotation bilinear resample, smA -> smB (reference pass 1)
    // 4 consecutive pixels per thread -> ds_store_b128
    for (int i = 0; i < 4; ++i) {
      int p0 = ((i << 10) + tid) << 2;       // base pixel, multiple of 4
      int h2 = p0 >> 7, w2 = p0 & HW_MASK;
      float4 v;
      v.x = rot_bilinear(smA, cA, sA, h2, w2);
      v.y = rot_bilinear(smA, cA, sA, h2, w2 + 1);
      v.z = rot_bilinear(smA, cA, sA, h2, w2 + 2);
      v.w = rot_bilinear(smA, cA, sA, h2, w2 + 3);
      *(float4*)(smB + p0) = v;
    }
    __syncthreads();                          // smA now dead for this channel

    // ---- overlap: async-stage next plane (channel c+1, or label plane)
    const float* nxt = (c < 2) ? (src0 + (size_t)(c + 1) * TILE) : lsrc;
    issue_stage_async(ldsA, nxt, tid);

    // ---- phase B: scale bilinear resample from smB with flip folded into
    // the gather indices (flip is a pixel permutation -> bit-exact).
    // y-geometry shared across the 4 pixels; global_store_b128 outputs.
    float* outc = out_p + (size_t)(n * 3 + c) * TILE;
    for (int i = 0; i < 4; ++i) {
      int p0 = ((i << 10) + tid) << 2;
      int h = p0 >> 7, w0 = p0 & HW_MASK;
      float Y = (2.0f * (float)h + 1.0f) * k - 1.0f;
      float iy = ((scl * Y + 1.0f) * 128.0f - 1.0f) * 0.5f;
      float y0 = floorf(iy);
      float wy1 = iy - y0, wy0 = 1.0f - wy1;
      float4 v;
      float* vp = &v.x;
#pragma unroll
      for (int j = 0; j < 4; ++j) {
        float X = (2.0f * (float)(w0 + j) + 1.0f) * k - 1.0f;
        float ix = ((scl * X + 1.0f) * 128.0f - 1.0f) * 0.5f;
        float x0 = floorf(ix);
        float wx1 = ix - x0, wx0 = 1.0f - wx1;
        float acc = 0.0f;
#pragma unroll
        for (int cy = 0; cy < 2; ++cy) {
          float yf = y0 + (float)cy;
          float wy = cy ? wy1 : wy0;
          bool inby = (yf >= 0.0f && yf <= 127.0f);
          int yc = (int)fminf(fmaxf(yf, 0.0f), 127.0f);
          int yrow = (yc ^ fvm) << 7;         // flipped intermediate row
#pragma unroll
          for (int cx = 0; cx < 2; ++cx) {
            float xf = x0 + (float)cx;
            float wx = cx ? wx1 : wx0;
            float inb = (inby && xf >= 0.0f && xf <= 127.0f) ? 1.0f : 0.0f;
            int xc = (int)fminf(fmaxf(xf, 0.0f), 127.0f);
            acc += smB[yrow + (xc ^ fhm)] * (wx * wy * inb);
          }
        }
        vp[j] = acc;
      }
      *(float4*)(outc + p0) = v;
    }
    wait_async_and_barrier();                 // next plane resident in smA
  }

  // ---- labels: nearest(scale) o flip o nearest(rot), fused index chase
  // from the staged label plane (now in smA). jnp.round == rintf (half-even).
  float* outl = out_l + (size_t)n * TILE;
  for (int i = 0; i < 4; ++i) {
    int p0 = ((i << 10) + tid) << 2;
    int h = p0 >> 7, w0 = p0 & HW_MASK;
    float Y = (2.0f * (float)h + 1.0f) * k - 1.0f;
    float iy = ((scl * Y + 1.0f) * 128.0f - 1.0f) * 0.5f;
    float yr = rintf(iy);
    bool inby = (yr >= 0.0f && yr <= 127.0f);
    int yc = (int)fminf(fmaxf(yr, 0.0f), 127.0f);
    int ys2 = yc ^ fvm;
    float Y2 = (2.0f * (float)ys2 + 1.0f) * k - 1.0f;
    float4 v;
    float* vp = &v.x;
#pragma unroll
    for (int j = 0; j < 4; ++j) {
      float X = (2.0f * (float)(w0 + j) + 1.0f) * k - 1.0f;
      float ix = ((scl * X + 1.0f) * 128.0f - 1.0f) * 0.5f;
      float xr = rintf(ix);
      float inb1 = (inby && xr >= 0.0f && xr <= 127.0f) ? 1.0f : 0.0f;
      int xc = (int)fminf(fmaxf(xr, 0.0f), 127.0f);
      int xs2 = xc ^ fhm;
      // rotation-nearest of the label at intermediate pixel (ys2, xs2)
      float X2 = (2.0f * (float)xs2 + 1.0f) * k - 1.0f;
      float gx2 = cA * X2 - sA * Y2;
      float gy2 = sA * X2 + cA * Y2;
      float jx = ((gx2 + 1.0f) * 128.0f - 1.0f) * 0.5f;
      float jy = ((gy2 + 1.0f) * 128.0f - 1.0f) * 0.5f;
      float xr2 = rintf(jx), yr2 = rintf(jy);
      float inb2 = (xr2 >= 0.0f && xr2 <= 127.0f && yr2 >= 0.0f && yr2 <= 127.0f)
                       ? 1.0f : 0.0f;
      int xc2 = (int)fminf(fmaxf(xr2, 0.0f), 127.0f);
      int yc2 = (int)fminf(fmaxf(yr2, 0.0f), 127.0f);
      vp[j] = (smA[(yc2 << 7) + xc2] * inb2) * inb1;
    }
    *(float4*)(outl + p0) = v;
  }
}

extern "C" void kernel_launch(void* const* d_in, const int* in_sizes, int n_in,
                              void* d_out, int out_size, void* d_ws, size_t ws_size,
                              hipStream_t stream) {
  const float*   patches = (const float*)d_in[0];   // (B,K,3,128,128) f32
  const float*   labels  = (const float*)d_in[1];   // (B,K,1,128,128) f32
  const float*   angles  = (const float*)d_in[2];   // (B,K) f32
  const uint8_t* fh      = (const uint8_t*)d_in[3]; // (B,K) bool (1 byte)
  const uint8_t* fv      = (const uint8_t*)d_in[4]; // (B,K) bool (1 byte)
  const float*   scales  = (const float*)d_in[5];   // (B,K) f32

  const int N = in_sizes[2];                        // B*K = 256 patches
  float* out_p = (float*)d_out;                     // first tuple element
  float* out_l = out_p + (size_t)N * 3 * TILE;      // second tuple element

  patch_augment_fused<<<N, 1024, 0, stream>>>(patches, labels, angles,
                                              fh, fv, scales, out_p, out_l);
}